// RotaryEmbeddingHilbert_70978629534158
// MI455X (gfx1250) — compile-verified
//

#include <hip/hip_runtime.h>
#include <stdint.h>

// Problem constants (reference: H = W = 256, DIM = 512, n = 256 -> all cells valid,
// rank == hilbert index, angle = d * freq, broadcast across 512 dims).
#define H_ 256
#define W_ 256
#define DIM_ 512
#define N_ (H_ * W_)

#define ROWS_PER_BLOCK 64
#define THREADS_ 256
// chunks of 16B per row: 512 floats * 4B / 16B = 128
#define CHUNKS_PER_ROW 128
#define CHUNKS_PER_BLOCK (ROWS_PER_BLOCK * CHUNKS_PER_ROW) // 8192
#define ITERS_ (CHUNKS_PER_BLOCK / THREADS_)               // 32

// ---- CDNA5 async LDS->global store path -------------------------------------
#define AS_GLOBAL __attribute__((address_space(1)))
#define AS_LDS __attribute__((address_space(3)))

typedef int v4i_ __attribute__((vector_size(4 * sizeof(int))));

#if __has_builtin(__builtin_amdgcn_global_store_async_from_lds_b128)
#define ASYNC_STORE_B128(gptr, lptr)                                           \
    __builtin_amdgcn_global_store_async_from_lds_b128(                         \
        (AS_GLOBAL v4i_*)(gptr), (AS_LDS v4i_*)(lptr), 0, 0)
#else
#define ASYNC_STORE_B128(gptr, lptr)                                           \
    asm volatile("global_store_async_from_lds_b128 %0, %1, off"                \
                 :                                                             \
                 : "v"((unsigned long long)(uintptr_t)(gptr)),                 \
                   "v"((unsigned)(size_t)(AS_LDS v4i_*)(lptr))                 \
                 : "memory")
#endif

#if __has_builtin(__builtin_amdgcn_s_wait_asynccnt)
#define WAIT_ASYNC0() __builtin_amdgcn_s_wait_asynccnt(0)
#else
#define WAIT_ASYNC0() asm volatile("s_wait_asynccnt 0" ::: "memory")
#endif
// -----------------------------------------------------------------------------

// Inverse of the reference's d2xy (verified against it for n=2 and n=4):
//   forward per level: rot(s) = [reflect if rx (ry==0)] then swap; this map is
//   its own inverse, so the standard top-down xy2d with bit-stripping matches.
__device__ __forceinline__ unsigned hilbert_xy2d_256(int x, int y) {
    unsigned d = 0;
#pragma unroll
    for (int s = 128; s > 0; s >>= 1) {
        int rx = (x & s) ? 1 : 0;
        int ry = (y & s) ? 1 : 0;
        d += (unsigned)(s * s) * (unsigned)((3 * rx) ^ ry);
        x &= (s - 1);
        y &= (s - 1);
        if (ry == 0) {
            int t = x;  // swap
            x = y;
            y = t;
            if (rx) {  // reflect within sub-square
                x = (s - 1) - x;
                y = (s - 1) - y;
            }
        }
    }
    return d;
}

__global__ void RotaryEmbeddingHilbert_70978629534158_kernel(
    const float* __restrict__ freq, float* __restrict__ out) {
    __shared__ float4 ldsv[ROWS_PER_BLOCK];  // one replicated 16B chunk per row

    const int tid = threadIdx.x;
    const int blk = blockIdx.x;

    // Phase 1: threads 0..63 compute one angle each, write replicated to LDS.
    if (tid < ROWS_PER_BLOCK) {
        const int p = blk * ROWS_PER_BLOCK + tid;  // output row = y*W + x
        const int y = p >> 8;
        const int x = p & (W_ - 1);
        const float a = (float)hilbert_xy2d_256(x, y) * freq[0];
        ldsv[tid] = make_float4(a, a, a, a);
    }
    __syncthreads();

    // Phase 2: fan LDS chunks out to global with async b128 stores.
    // Block region: 64 rows * 2048 B = 128 KiB contiguous.
    float* const base = out + (size_t)blk * (ROWS_PER_BLOCK * DIM_);
#pragma unroll 4
    for (int i = 0; i < ITERS_; ++i) {
        const int g = (i << 8) + tid;   // 16B-chunk index within block region
        const int r = g >> 7;           // source row in LDS (128 chunks/row)
        float* const gptr = base + ((size_t)g << 2);
        ASYNC_STORE_B128(gptr, &ldsv[r]);
    }
    WAIT_ASYNC0();
}

extern "C" void kernel_launch(void* const* d_in, const int* in_sizes, int n_in,
                              void* d_out, int out_size, void* d_ws, size_t ws_size,
                              hipStream_t stream) {
    const float* freq = (const float*)d_in[0];  // single learned scalar
    float* out = (float*)d_out;                 // 65536 * 512 fp32

    dim3 grid(N_ / ROWS_PER_BLOCK);  // 1024 blocks
    dim3 block(THREADS_);            // 256 threads = 8 wave32
    RotaryEmbeddingHilbert_70978629534158_kernel<<<grid, block, 0, stream>>>(freq, out);
}